// LQRStep_81312320848502
// MI455X (gfx1250) — compile-verified
//
#include <hip/hip_runtime.h>

// LQR solve for MI455X (gfx1250, wave32).
// Backward Riccati recursion: one 256-thread workgroup per batch element,
// V (64x64) carried in LDS across the 50 sequential steps; all large GEMMs
// done with V_WMMA_F32_16X16X4_F32 (full fp32 to keep Q_uu SPD for Cholesky).

#define NSZ 64   // n_state
#define NCZ 32   // n_ctrl
#define NNZ 96   // n_state + n_ctrl
#define TSZ 50
#define BSZ 256

typedef __attribute__((ext_vector_type(2))) float v2f;
typedef __attribute__((ext_vector_type(8))) float v8f;

// ---------------------------------------------------------------------------
// One 16x16 output tile, D += A(16xK) * B(Kx16), fp32 WMMA, K multiple of 4.
// A layout (ISA 7.12.2, 32-bit A 16x4): lanes 0-15 hold K=kk,kk+1 for M=lane;
// lanes 16-31 hold K=kk+2,kk+3.  B: row K striped across lanes (N=lane&15),
// mirrored halves.  AT=true reads A as transposed (A[k][m]).
// ---------------------------------------------------------------------------
template <bool AT>
__device__ __forceinline__ v8f wmma_acc_tile(const float* A, int lda,
                                             const float* B, int ldb,
                                             int mBase, int nBase, int kLen,
                                             v8f acc, int lane) {
  const int half = lane >> 4;
  const int l16  = lane & 15;
  const int m = mBase + l16;
  const int n = nBase + l16;
#pragma unroll 4
  for (int kk = 0; kk < kLen; kk += 4) {
    const int k0 = kk + half * 2;
    v2f a, b;
    if (AT) {
      a.x = A[(size_t)k0 * lda + m];
      a.y = A[(size_t)(k0 + 1) * lda + m];
    } else {
      a.x = A[(size_t)m * lda + k0];
      a.y = A[(size_t)m * lda + k0 + 1];
    }
    b.x = B[(size_t)k0 * ldb + n];
    b.y = B[(size_t)(k0 + 1) * ldb + n];
    acc = __builtin_amdgcn_wmma_f32_16x16x4_f32(false, a, false, b,
                                                (short)0, acc, false, false);
  }
  return acc;
}

// C/D layout (ISA 7.12.2): VGPR r, lanes 0-15 -> M=r; lanes 16-31 -> M=8+r.
__device__ __forceinline__ v8f load_tile(const float* S, int lds_,
                                         int mBase, int nBase, int lane) {
  const int half = lane >> 4, l16 = lane & 15;
  v8f acc;
#pragma unroll
  for (int r = 0; r < 8; ++r)
    acc[r] = S[(size_t)(mBase + half * 8 + r) * lds_ + nBase + l16];
  return acc;
}

__device__ __forceinline__ void store_tile(float* D, int ldd,
                                           int mBase, int nBase, v8f acc, int lane) {
  const int half = lane >> 4, l16 = lane & 15;
#pragma unroll
  for (int r = 0; r < 8; ++r)
    D[(size_t)(mBase + half * 8 + r) * ldd + nBase + l16] = acc[r];
}

// ---------------------------------------------------------------------------
// Backward Riccati sweep.  grid = B blocks of 256 threads (8 waves).
// ---------------------------------------------------------------------------
__global__ __launch_bounds__(256) void lqr_backward(
    const float* __restrict__ C,   // (T,B,96,96)
    const float* __restrict__ c,   // (T,B,96)
    const float* __restrict__ F,   // (T,B,64,96)
    const float* __restrict__ cx,  // (T,B,64)
    const float* __restrict__ cu,  // (T,B,32)
    float* __restrict__ Kw,        // (T,B,32,64) workspace
    float* __restrict__ kw)        // (T,B,32)    workspace
{
  extern __shared__ float smem[];
  float* sV   = smem;                 // 64x64
  float* sF   = sV + NSZ * NSZ;       // 64x96
  float* sM   = sF + NSZ * NNZ;       // 64x96  (M = V*F)
  float* sQ   = sM + NSZ * NNZ;       // 96x96
  float* sL   = sQ + NNZ * NNZ;       // 32x32 Cholesky factor
  float* sK   = sL + NCZ * NCZ;       // 32x64 feedback gain
  float* sW   = sK + NCZ * NSZ;       // 32x64 (W = Quu*K)
  float* sxut = sW + NCZ * NSZ;       // 96
  float* sq   = sxut + NNZ;           // 96
  float* sv   = sq + NNZ;             // 64
  float* skv  = sv + NSZ;             // 32
  float* sw2  = skv + NCZ;            // 32

  const int b    = blockIdx.x;
  const int tid  = threadIdx.x;
  const int lane = tid & 31;
  const int wave = tid >> 5;

  for (int i = tid; i < NSZ * NSZ; i += 256) sV[i] = 0.0f;
  if (tid < NSZ) sv[tid] = 0.0f;
  __syncthreads();

  for (int t = TSZ - 1; t >= 0; --t) {
    const size_t tb = (size_t)t * BSZ + b;
    const float* Ct = C + tb * NNZ * NNZ;
    const float* Ft = F + tb * NSZ * NNZ;
    const float* ct = c + tb * NNZ;

    // prefetch next step's F and C while this step computes
    if (t > 0) {
      const size_t nb = (size_t)(t - 1) * BSZ + b;
      __builtin_prefetch(F + nb * NSZ * NNZ + (size_t)tid * 24, 0, 0);
      __builtin_prefetch(C + nb * NNZ * NNZ + (size_t)tid * 36, 0, 0);
    }

    // ---- phase 1: stage F and xut ------------------------------------
    for (int i = tid; i < NSZ * NNZ; i += 256) sF[i] = Ft[i];
    if (tid < NSZ)          sxut[tid] = cx[tb * NSZ + tid];
    else if (tid < NNZ)     sxut[tid] = cu[tb * NCZ + (tid - NSZ)];
    __syncthreads();

    // ---- phase 2: M = V * F   (24 tiles, K=64) -----------------------
    for (int tile = wave; tile < 24; tile += 8) {
      const int tm = (tile / 6) * 16, tn = (tile % 6) * 16;
      v8f acc = {};
      acc = wmma_acc_tile<false>(sV, NSZ, sF, NNZ, tm, tn, NSZ, acc, lane);
      store_tile(sM, NNZ, tm, tn, acc, lane);
    }
    __syncthreads();

    // ---- phase 3: Q = C + F^T * M  (36 tiles) + q vector -------------
    for (int tile = wave; tile < 36; tile += 8) {
      const int tm = (tile / 6) * 16, tn = (tile % 6) * 16;
      v8f acc = load_tile(Ct, NNZ, tm, tn, lane);
      acc = wmma_acc_tile<true>(sF, NNZ, sM, NNZ, tm, tn, NSZ, acc, lane);
      store_tile(sQ, NNZ, tm, tn, acc, lane);
    }
    // q_j = c_j + sum_i C[j][i]*xut_i + sum_i F[i][j]*v_i   (waves 0..2)
    if (tid < NNZ) {
      float s = ct[tid];
      for (int j = 0; j < NNZ; ++j) s += Ct[(size_t)tid * NNZ + j] * sxut[j];
      for (int i = 0; i < NSZ; ++i) s += sF[(size_t)i * NNZ + tid] * sv[i];
      sq[tid] = s;
    }
    __syncthreads();

    // ---- phase 4: Cholesky of Q_uu (32x32 SPD) -----------------------
    if (tid < NCZ * NCZ)
      sL[tid] = sQ[(size_t)(NSZ + tid / NCZ) * NNZ + NSZ + (tid % NCZ)];
    __syncthreads();
    for (int kc = 0; kc < NCZ; ++kc) {
      if (tid == 0) sL[kc * NCZ + kc] = sqrtf(sL[kc * NCZ + kc]);
      __syncthreads();
      if (tid > kc && tid < NCZ) sL[tid * NCZ + kc] /= sL[kc * NCZ + kc];
      __syncthreads();
      const int rem = NCZ - kc - 1;
      for (int idx = tid; idx < rem * rem; idx += 256) {
        const int i = kc + 1 + idx / rem;
        const int j = kc + 1 + idx % rem;
        if (j <= i) sL[i * NCZ + j] -= sL[i * NCZ + kc] * sL[j * NCZ + kc];
      }
      __syncthreads();
    }

    // ---- phase 5: solve Quu X = [Q_ux | q_u], K = -X, k = -x ---------
    if (tid < NSZ + 1) {
      float y[NCZ];
      for (int i = 0; i < NCZ; ++i)
        y[i] = (tid < NSZ) ? sQ[(size_t)(NSZ + i) * NNZ + tid] : sq[NSZ + i];
      for (int i = 0; i < NCZ; ++i) {          // forward subst (L)
        float s = y[i];
        for (int j = 0; j < i; ++j) s -= sL[i * NCZ + j] * y[j];
        y[i] = s / sL[i * NCZ + i];
      }
      for (int i = NCZ - 1; i >= 0; --i) {     // backward subst (L^T)
        float s = y[i];
        for (int j = i + 1; j < NCZ; ++j) s -= sL[j * NCZ + i] * y[j];
        y[i] = s / sL[i * NCZ + i];
      }
      if (tid < NSZ) { for (int i = 0; i < NCZ; ++i) sK[i * NSZ + tid] = -y[i]; }
      else           { for (int i = 0; i < NCZ; ++i) skv[i]            = -y[i]; }
    }
    __syncthreads();

    // ---- phase 6: W = Quu * K (8 tiles) and w2 = Quu * k -------------
    {
      const int tile = wave;
      const int tm = (tile / 4) * 16, tn = (tile % 4) * 16;
      v8f acc = {};
      acc = wmma_acc_tile<false>(sQ + (size_t)NSZ * NNZ + NSZ, NNZ,
                                 sK, NSZ, tm, tn, NCZ, acc, lane);
      store_tile(sW, NSZ, tm, tn, acc, lane);
    }
    if (tid < NCZ) {
      float s = 0.0f;
      for (int k2 = 0; k2 < NCZ; ++k2)
        s += sQ[(size_t)(NSZ + tid) * NNZ + NSZ + k2] * skv[k2];
      sw2[tid] = s;
    }
    __syncthreads();

    // ---- phase 7: Vn = Qxx + Qxu K + K^T Qux + K^T W  -> sV ----------
    for (int tile = wave; tile < 16; tile += 8) {
      const int tm = (tile / 4) * 16, tn = (tile % 4) * 16;
      v8f acc = load_tile(sQ, NNZ, tm, tn, lane);                        // Q_xx
      acc = wmma_acc_tile<false>(sQ + NSZ, NNZ, sK, NSZ, tm, tn, NCZ, acc, lane); // Qxu*K
      acc = wmma_acc_tile<true>(sK, NSZ, sQ + (size_t)NSZ * NNZ, NNZ,
                                tm, tn, NCZ, acc, lane);                 // K^T*Qux
      acc = wmma_acc_tile<true>(sK, NSZ, sW, NSZ, tm, tn, NCZ, acc, lane); // K^T*W
      store_tile(sV, NSZ, tm, tn, acc, lane);
    }
    // vn = q_x + Qxu k + K^T q_u + K^T (Quu k)
    if (tid < NSZ) {
      float s = sq[tid];
      for (int k2 = 0; k2 < NCZ; ++k2) s += sQ[(size_t)tid * NNZ + NSZ + k2] * skv[k2];
      for (int k2 = 0; k2 < NCZ; ++k2) s += sK[k2 * NSZ + tid] * sq[NSZ + k2];
      for (int k2 = 0; k2 < NCZ; ++k2) s += sK[k2 * NSZ + tid] * sw2[k2];
      sv[tid] = s;
    }
    // spill K, k for the forward rollout
    {
      float* Kt = Kw + tb * NCZ * NSZ;
      for (int i = tid; i < NCZ * NSZ; i += 256) Kt[i] = sK[i];
      if (tid < NCZ) kw[tb * NCZ + tid] = skv[tid];
    }
    __syncthreads();
  }
}

// ---------------------------------------------------------------------------
// Forward rollout.  grid = B blocks of 128 threads.  Pure mat-vec.
// ---------------------------------------------------------------------------
__global__ __launch_bounds__(128) void lqr_forward(
    const float* __restrict__ F, const float* __restrict__ cx,
    const float* __restrict__ cu, const float* __restrict__ x_init,
    const float* __restrict__ Kw, const float* __restrict__ kw,
    float* __restrict__ out_x, float* __restrict__ out_u)
{
  __shared__ float sx[NSZ], sdx[NSZ], su[NCZ], sxut[NNZ];
  const int b = blockIdx.x, tid = threadIdx.x;
  if (tid < NSZ) { sx[tid] = x_init[(size_t)b * NSZ + tid]; sdx[tid] = 0.0f; }
  __syncthreads();

  for (int t = 0; t < TSZ; ++t) {
    const size_t tb = (size_t)t * BSZ + b;
    const float* Kt = Kw + tb * NCZ * NSZ;
    const float* kt = kw + tb * NCZ;
    const float* Ft = F + tb * NSZ * NNZ;

    if (tid < NCZ) {                       // new_u = K dx + u + k
      float s = cu[tb * NCZ + tid] + kt[tid];
      for (int j = 0; j < NSZ; ++j) s += Kt[(size_t)tid * NSZ + j] * sdx[j];
      su[tid] = s;
    }
    __syncthreads();

    if (tid < NSZ) out_x[tb * NSZ + tid] = sx[tid];
    if (tid < NCZ) out_u[tb * NCZ + tid] = su[tid];
    if (tid < NSZ)      sxut[tid] = sx[tid];
    else if (tid < NNZ) sxut[tid] = su[tid - NSZ];
    __syncthreads();

    if (tid < NSZ) {                       // x_next = F [x; new_u]
      float s = 0.0f;
      for (int j = 0; j < NNZ; ++j) s += Ft[(size_t)tid * NNZ + j] * sxut[j];
      const float xr = (t + 1 < TSZ) ? cx[((size_t)(t + 1) * BSZ + b) * NSZ + tid] : 0.0f;
      sx[tid] = s;
      sdx[tid] = s - xr;
    }
    __syncthreads();
  }
}

// ---------------------------------------------------------------------------
extern "C" void kernel_launch(void* const* d_in, const int* in_sizes, int n_in,
                              void* d_out, int out_size, void* d_ws, size_t ws_size,
                              hipStream_t stream) {
  (void)in_sizes; (void)n_in; (void)out_size; (void)ws_size;
  const float* x_init = (const float*)d_in[0];
  const float* C      = (const float*)d_in[1];
  const float* c      = (const float*)d_in[2];
  const float* F      = (const float*)d_in[3];
  const float* curx   = (const float*)d_in[4];
  const float* curu   = (const float*)d_in[5];

  float* Kw = (float*)d_ws;                                   // T*B*32*64
  float* kw = Kw + (size_t)TSZ * BSZ * NCZ * NSZ;             // T*B*32
  float* out_x = (float*)d_out;                               // T*B*64
  float* out_u = out_x + (size_t)TSZ * BSZ * NSZ;             // T*B*32

  const size_t shmem = (size_t)(NSZ*NSZ + 2*NSZ*NNZ + NNZ*NNZ + NCZ*NCZ +
                                2*NCZ*NSZ + 2*NNZ + NSZ + 2*NCZ + NSZ) * sizeof(float);

  lqr_backward<<<BSZ, 256, shmem, stream>>>(C, c, F, curx, curu, Kw, kw);
  lqr_forward<<<BSZ, 128, 0, stream>>>(F, curx, curu, x_init, Kw, kw, out_x, out_u);
}